// DisSimilarity_Loss_Efficiency_46334107189857
// MI455X (gfx1250) — compile-verified
//
#include <hip/hip_runtime.h>
#include <math.h>

// Problem shape (fixed by the reference): z_list (P,B,D) fp32.
#define PP 64
#define BB 512
#define DD 1024
#define EPSV 1e-8f

// ---------------- CDNA5 async global->LDS path ------------------------------
typedef int v4i __attribute__((ext_vector_type(4)));
typedef __attribute__((address_space(1))) v4i* gptr_v4i;  // global
typedef __attribute__((address_space(3))) v4i* lptr_v4i;  // LDS

#if defined(__has_builtin)
#  if __has_builtin(__builtin_amdgcn_global_load_async_to_lds_b128)
#    define HAVE_ASYNC_B128 1
#  endif
#  if __has_builtin(__builtin_amdgcn_s_wait_asynccnt)
#    define HAVE_WAIT_ASYNC 1
#  endif
#endif

__device__ __forceinline__ void async_copy_b128(const float* gsrc, float* lds_dst) {
#if defined(HAVE_ASYNC_B128)
  __builtin_amdgcn_global_load_async_to_lds_b128(
      (gptr_v4i)(void*)const_cast<float*>(gsrc),
      (lptr_v4i)(void*)lds_dst,
      /*ioffset=*/0, /*cpol=*/0);
#else
  unsigned lds_off = (unsigned)(size_t)(__attribute__((address_space(3))) void*)lds_dst;
  asm volatile("global_load_async_to_lds_b128 %0, %1, off"
               :: "v"(lds_off), "v"(gsrc) : "memory");
#endif
}

#if defined(HAVE_WAIT_ASYNC)
#  define WAIT_ASYNCCNT(N) __builtin_amdgcn_s_wait_asynccnt(N)
#else
#  define WAIT_ASYNCCNT(N) asm volatile("s_wait_asynccnt " #N ::: "memory")
#endif

// ---------------- reductions ------------------------------------------------
__device__ __forceinline__ float waveReduceAdd(float v) {  // wave32
  v += __shfl_xor(v, 16, 32);
  v += __shfl_xor(v, 8, 32);
  v += __shfl_xor(v, 4, 32);
  v += __shfl_xor(v, 2, 32);
  v += __shfl_xor(v, 1, 32);
  return v;
}

// ---------------------------------------------------------------------------
// Kernel 1: one workgroup per b. Streams the 64 rows z_list[p,b,:] (4 KB each,
// 2 MB apart) through a double-buffered LDS stage via async loads. Produces
// Q[b,:] = sum_p zn[p,b,:] and AN[b,:] = normalize(mean_p z_list[p,b,:]).
// Thread t owns d in [4t, 4t+4): each lane consumes exactly the 16 bytes its
// own async op wrote, so the staging buffers need no cross-wave barriers.
// ---------------------------------------------------------------------------
__global__ __launch_bounds__(256) void k_rows(const float* __restrict__ z,
                                              float* __restrict__ Q,
                                              float* __restrict__ AN) {
  const int b    = blockIdx.x;
  const int tid  = threadIdx.x;
  const int lane = tid & 31;
  const int wid  = tid >> 5;

  __shared__ __align__(16) float buf[2][DD];   // 8 KB double buffer
  __shared__ float red[2][8];                  // per-row norm reduction (8 waves)
  __shared__ float red2[8];                    // z_avg norm reduction

  const float* gbase = z + (size_t)b * DD + (size_t)tid * 4;

  // prologue: prefetch row p=0
  async_copy_b128(gbase, &buf[0][tid * 4]);

  float as0 = 0.f, as1 = 0.f, as2 = 0.f, as3 = 0.f;  // sum_p z
  float aq0 = 0.f, aq1 = 0.f, aq2 = 0.f, aq3 = 0.f;  // sum_p zn

  for (int p = 0; p < PP; ++p) {
    if (p + 1 < PP) {
      async_copy_b128(gbase + (size_t)(p + 1) * ((size_t)BB * DD),
                      &buf[(p + 1) & 1][tid * 4]);
      WAIT_ASYNCCNT(1);   // oldest (row p) has landed; prefetch p+1 in flight
    } else {
      WAIT_ASYNCCNT(0);
    }

    const float4 v = *(const float4*)&buf[p & 1][tid * 4];

    float ss = v.x * v.x + v.y * v.y + v.z * v.z + v.w * v.w;
    ss = waveReduceAdd(ss);
    if (lane == 0) red[p & 1][wid] = ss;
    __syncthreads();
    const float* r = red[p & 1];
    float tot = r[0] + r[1] + r[2] + r[3] + r[4] + r[5] + r[6] + r[7];
    float inv = 1.0f / fmaxf(sqrtf(tot), EPSV);

    as0 += v.x;       as1 += v.y;       as2 += v.z;       as3 += v.w;
    aq0 += v.x * inv; aq1 += v.y * inv; aq2 += v.z * inv; aq3 += v.w * inv;
  }

  // z_avg row = (sum_p z)/P ; an = z_avg / max(||z_avg||, eps)
  const float invP = 1.0f / (float)PP;
  float a0 = as0 * invP, a1 = as1 * invP, a2 = as2 * invP, a3 = as3 * invP;
  float ss2 = a0 * a0 + a1 * a1 + a2 * a2 + a3 * a3;
  ss2 = waveReduceAdd(ss2);
  __syncthreads();
  if (lane == 0) red2[wid] = ss2;
  __syncthreads();
  float tot2 = red2[0] + red2[1] + red2[2] + red2[3] +
               red2[4] + red2[5] + red2[6] + red2[7];
  float inv2 = 1.0f / fmaxf(sqrtf(tot2), EPSV);

  float4 qv = make_float4(aq0, aq1, aq2, aq3);
  float4 av = make_float4(a0 * inv2, a1 * inv2, a2 * inv2, a3 * inv2);
  *(float4*)&Q [(size_t)b * DD + (size_t)tid * 4] = qv;
  *(float4*)&AN[(size_t)b * DD + (size_t)tid * 4] = av;
}

// ---------------------------------------------------------------------------
// Kernel 2: column reduce over b. S[d]=sum_b Q[b,d], T[d]=sum_b AN[b,d],
// PD[d]=sum_b Q[b,d]*AN[b,d] (the diagonal term). Coalesced reads.
// ---------------------------------------------------------------------------
__global__ __launch_bounds__(256) void k_cols(const float* __restrict__ Q,
                                              const float* __restrict__ AN,
                                              float* __restrict__ S,
                                              float* __restrict__ T,
                                              float* __restrict__ PD) {
  const int d = blockIdx.x * 256 + threadIdx.x;  // grid = DD/256
  float s = 0.f, t = 0.f, pd = 0.f;
  for (int b = 0; b < BB; ++b) {
    float q = Q [(size_t)b * DD + d];
    float a = AN[(size_t)b * DD + d];
    s += q; t += a; pd += q * a;
  }
  S[d] = s; T[d] = t; PD[d] = pd;
}

// ---------------------------------------------------------------------------
// Kernel 3: result = (S.T - diag)/count - 1,  count = P*B*(B-1)
// ---------------------------------------------------------------------------
__global__ __launch_bounds__(256) void k_final(const float* __restrict__ S,
                                               const float* __restrict__ T,
                                               const float* __restrict__ PD,
                                               float* __restrict__ out) {
  const int tid = threadIdx.x;
  float st = 0.f, dg = 0.f;
  for (int d = tid; d < DD; d += 256) { st += S[d] * T[d]; dg += PD[d]; }
  __shared__ float r1[8], r2[8];
  st = waveReduceAdd(st);
  dg = waveReduceAdd(dg);
  if ((tid & 31) == 0) { r1[tid >> 5] = st; r2[tid >> 5] = dg; }
  __syncthreads();
  if (tid == 0) {
    float sst = 0.f, sdg = 0.f;
    for (int i = 0; i < 8; ++i) { sst += r1[i]; sdg += r2[i]; }
    const double count = (double)PP * (double)BB * (double)(BB - 1);
    out[0] = (float)(((double)sst - (double)sdg) / count - 1.0);
  }
}

// ---------------------------------------------------------------------------
extern "C" void kernel_launch(void* const* d_in, const int* in_sizes, int n_in,
                              void* d_out, int out_size, void* d_ws, size_t ws_size,
                              hipStream_t stream) {
  const float* z = (const float*)d_in[0];   // z_list (P,B,D) fp32; d_in[1] unused
  float* ws = (float*)d_ws;
  // ws layout (floats): Q[B*D] | AN[B*D] | S[D] | T[D] | PD[D]  (~4.2 MB)
  float* Q  = ws;
  float* AN = ws + (size_t)BB * DD;
  float* S  = ws + 2 * (size_t)BB * DD;
  float* T  = S + DD;
  float* PD = T + DD;

  k_rows<<<BB, 256, 0, stream>>>(z, Q, AN);
  k_cols<<<DD / 256, 256, 0, stream>>>(Q, AN, S, T, PD);
  k_final<<<1, 256, 0, stream>>>(S, T, PD, (float*)d_out);
}